// BoxPool_69741678952496
// MI455X (gfx1250) — compile-verified
//
#include <hip/hip_runtime.h>
#include <stdint.h>

#define BB 8
#define NN 2134
#define CC 81
#define WW ((NN + 31) / 32)   /* 67 mask words per column */
#define KK 32                 /* neighbor-list cap */
#define THR 0.7f

// ---------------------------------------------------------------------------
// Phase 1: per-batch pairwise IoU threshold mask, column-major bitwords.
// mask[(b*WW + w)*NN + j] bit k  <=>  IoU(row i = w*32+k, col j) >= 0.7
// ---------------------------------------------------------------------------
__global__ __launch_bounds__(256) void iou_mask_kernel(
    const float* __restrict__ box, uint32_t* __restrict__ mask) {
  const int b = blockIdx.z;
  const int w = blockIdx.y;
  const int j = blockIdx.x * 256 + threadIdx.x;

  __shared__ float rx1[32], ry1[32], rx2[32], ry2[32], rar[32];
  const int i0 = w * 32;
  if (threadIdx.x < 32) {
    const int i = i0 + threadIdx.x;
    float x1 = 0.f, y1 = 0.f, x2 = 0.f, y2 = 0.f;
    if (i < NN) {
      x1 = box[(b * 4 + 0) * NN + i];
      y1 = box[(b * 4 + 1) * NN + i];
      x2 = box[(b * 4 + 2) * NN + i];
      y2 = box[(b * 4 + 3) * NN + i];
    }
    rx1[threadIdx.x] = x1; ry1[threadIdx.x] = y1;
    rx2[threadIdx.x] = x2; ry2[threadIdx.x] = y2;
    rar[threadIdx.x] = (x2 - x1) * (y2 - y1);
  }
  __syncthreads();
  if (j >= NN) return;

  const float bx1 = box[(b * 4 + 0) * NN + j];
  const float by1 = box[(b * 4 + 1) * NN + j];
  const float bx2 = box[(b * 4 + 2) * NN + j];
  const float by2 = box[(b * 4 + 3) * NN + j];
  const float areaB = (bx2 - bx1) * (by2 - by1);

  uint32_t bits = 0u;
  int kmax = NN - i0; if (kmax > 32) kmax = 32;
  for (int k = 0; k < kmax; ++k) {
    float iw = fminf(rx2[k], bx2) - fmaxf(rx1[k], bx1);
    float ih = fminf(ry2[k], by2) - fmaxf(ry1[k], by1);
    iw = fmaxf(iw, 0.0f);
    ih = fmaxf(ih, 0.0f);
    const float inter = iw * ih;
    const float iou = inter / (rar[k] + areaB - inter);
    if (iou >= THR) bits |= (1u << k);
  }
  mask[((size_t)b * WW + w) * NN + j] = bits;
}

// ---------------------------------------------------------------------------
// Phase 1.5: compact per-column neighbor lists (class-independent).
// cnt[b*NN+j] = #masked i != j; indices stored if count <= KK, else
// phase 2 falls back to scanning the bitmask for that column.
// ---------------------------------------------------------------------------
__global__ __launch_bounds__(256) void build_lists_kernel(
    const uint32_t* __restrict__ mask, uint32_t* __restrict__ cnt,
    uint32_t* __restrict__ nbr) {
  const int t = blockIdx.x * 256 + threadIdx.x;
  if (t >= BB * NN) return;
  const int b = t / NN;
  const int j = t - b * NN;
  const uint32_t* mcol = mask + (size_t)b * WW * NN + j;
  uint32_t c = 0;
  for (int w = 0; w < WW; ++w) {
    uint32_t word = mcol[(size_t)w * NN];
    while (word) {
      const int k = __builtin_ctz(word);
      word &= word - 1u;
      const int i = w * 32 + k;
      if (i == j) continue;
      if (c < KK) nbr[(size_t)t * KK + c] = (uint32_t)i;
      ++c;
    }
  }
  cnt[t] = c;
}

// ---------------------------------------------------------------------------
// Phase 2: one block per (b, class). The class-score row (8.5 KB) is staged
// into LDS exactly once per (b,c) with gfx1250 async B64 copies (rows are
// 8-byte aligned); each thread then resolves ~NN/256 columns from the
// neighbor lists with exact jnp.argmax first-max tie semantics.
// ---------------------------------------------------------------------------
__global__ __launch_bounds__(256) void pool_kernel(
    const float* __restrict__ score, const uint32_t* __restrict__ mask,
    const uint32_t* __restrict__ cnt, const uint32_t* __restrict__ nbr,
    float* __restrict__ out) {
  const int c = blockIdx.x;
  const int b = blockIdx.y;
  const size_t outBase = ((size_t)b * CC + c) * NN;

  if (c == 0) {  // background class: all ones
    for (int j = threadIdx.x; j < NN; j += 256) out[outBase + j] = 1.0f;
    return;
  }

  __shared__ __align__(16) float s_lds[NN];
  const float* __restrict__ srow = score + ((size_t)b * CC + c) * NN;

  // CDNA5 async copy global->LDS (ASYNCcnt). NN = 2*1067 dwords exactly.
  {
    const unsigned long long sbase = (unsigned long long)(uintptr_t)srow;
    const uint32_t ldsBase = (uint32_t)(uintptr_t)(&s_lds[0]);
    for (int v = threadIdx.x; v < NN / 2; v += 256) {
      const uint32_t ldsOff = ldsBase + (uint32_t)(v * 8);
      const uint32_t gOff = (uint32_t)(v * 8);
      asm volatile("global_load_async_to_lds_b64 %0, %1, %2"
                   :
                   : "v"(ldsOff), "v"(gOff), "s"(sbase)
                   : "memory");
    }
    asm volatile("s_wait_asynccnt 0" ::: "memory");
  }
  __syncthreads();

  for (int j = threadIdx.x; j < NN; j += 256) {
    const float sj = s_lds[j];
    const int bn = b * NN + j;
    bool sup = false;

    const uint32_t cj = cnt[bn];
    if (cj <= KK) {
      const uint32_t* __restrict__ lst = nbr + (size_t)bn * KK;
      for (uint32_t t = 0; t < cj; ++t) {
        const int i = (int)lst[t];
        const float si = s_lds[i];
        if (si > sj || (si == sj && i < j)) { sup = true; break; }
      }
    } else {  // overflow fallback: scan bitmask column
      const uint32_t* __restrict__ mcol = mask + (size_t)b * WW * NN + j;
      for (int w = 0; w < WW && !sup; ++w) {
        uint32_t word = mcol[(size_t)w * NN];
        while (word) {
          const int k = __builtin_ctz(word);
          word &= word - 1u;
          const int i = w * 32 + k;
          if (i == j) continue;
          const float si = s_lds[i];
          if (si > sj || (si == sj && i < j)) { sup = true; break; }
        }
      }
    }
    // jnp.argmax corner case: s_j == 0 with j > 0 -> some earlier zero entry
    // (masked or not) ties at the max, so argmax < j.
    if (sj == 0.0f && j > 0) sup = true;
    out[outBase + j] = sup ? 0.0f : 1.0f;
  }
}

// ---------------------------------------------------------------------------
// No-workspace fused fallback (used only if ws_size is too small).
// One block per (b, class); boxes + scores staged in LDS once.
// ---------------------------------------------------------------------------
__global__ __launch_bounds__(256) void fused_kernel(
    const float* __restrict__ box, const float* __restrict__ score,
    float* __restrict__ out) {
  const int c = blockIdx.x;
  const int b = blockIdx.y;
  const size_t outBase = ((size_t)b * CC + c) * NN;
  if (c == 0) {
    for (int j = threadIdx.x; j < NN; j += 256) out[outBase + j] = 1.0f;
    return;
  }

  __shared__ float s_lds[NN];
  __shared__ float x1s[NN], y1s[NN], x2s[NN], y2s[NN];
  const float* __restrict__ srow = score + ((size_t)b * CC + c) * NN;
  const float* __restrict__ brow = box + (size_t)b * 4 * NN;
  for (int idx = threadIdx.x; idx < NN; idx += 256) {
    s_lds[idx] = srow[idx];
    x1s[idx] = brow[0 * NN + idx];
    y1s[idx] = brow[1 * NN + idx];
    x2s[idx] = brow[2 * NN + idx];
    y2s[idx] = brow[3 * NN + idx];
  }
  __syncthreads();

  for (int j = threadIdx.x; j < NN; j += 256) {
    const float bx1 = x1s[j], by1 = y1s[j], bx2 = x2s[j], by2 = y2s[j];
    const float areaB = (bx2 - bx1) * (by2 - by1);
    const float sj = s_lds[j];
    bool sup = false;
    for (int i = 0; i < NN; ++i) {
      if (i == j) continue;
      const float si = s_lds[i];
      const bool beats = (si > sj) || (si == sj && i < j);
      if (!beats) continue;
      float iw = fminf(x2s[i], bx2) - fmaxf(x1s[i], bx1);
      float ih = fminf(y2s[i], by2) - fmaxf(y1s[i], by1);
      iw = fmaxf(iw, 0.0f);
      ih = fmaxf(ih, 0.0f);
      const float inter = iw * ih;
      const float areaA = (x2s[i] - x1s[i]) * (y2s[i] - y1s[i]);
      if (inter / (areaA + areaB - inter) >= THR) { sup = true; break; }
    }
    if (sj == 0.0f && j > 0) sup = true;
    out[outBase + j] = sup ? 0.0f : 1.0f;
  }
}

// ---------------------------------------------------------------------------
extern "C" void kernel_launch(void* const* d_in, const int* in_sizes, int n_in,
                              void* d_out, int out_size, void* d_ws, size_t ws_size,
                              hipStream_t stream) {
  (void)in_sizes; (void)n_in; (void)out_size;
  const float* box = (const float*)d_in[0];    // [B,4,N] f32
  const float* score = (const float*)d_in[1];  // [B,C,N] f32
  float* out = (float*)d_out;                  // [B,C,N]

  const size_t maskWords = (size_t)BB * WW * NN;
  const size_t cntWords = (size_t)BB * NN;
  const size_t nbrWords = (size_t)BB * NN * KK;
  const size_t needBytes = (maskWords + cntWords + nbrWords) * sizeof(uint32_t);

  if (ws_size >= needBytes) {
    uint32_t* mask = (uint32_t*)d_ws;
    uint32_t* cnt = mask + maskWords;
    uint32_t* nbr = cnt + cntWords;

    dim3 g1((NN + 255) / 256, WW, BB);
    iou_mask_kernel<<<g1, 256, 0, stream>>>(box, mask);

    build_lists_kernel<<<(BB * NN + 255) / 256, 256, 0, stream>>>(mask, cnt, nbr);

    dim3 g2(CC, BB);
    pool_kernel<<<g2, 256, 0, stream>>>(score, mask, cnt, nbr, out);
  } else {
    dim3 g(CC, BB);
    fused_kernel<<<g, 256, 0, stream>>>(box, score, out);
  }
}